// HeteroscedasticCrossEntropy_28698971472081
// MI455X (gfx1250) — compile-verified
//
#include <hip/hip_runtime.h>

// ---------------------------------------------------------------------------
// Heteroscedastic MC cross-entropy, MI455X (gfx1250, wave32).
// B=8, C=19, H=W=256, S=100 MC samples. Output: single f32 scalar.
// TRANS/VALU-bound (~1e9 v_exp_f32). Per-pixel state register-resident.
// Log2-domain softmax (raw v_exp_f32/v_log_f32), V_DOT4_U32_U8 CLT Gaussian,
// V_WMMA_F32_16X16X4_F32 wave reduction, global_prefetch_b8 warmup.
// ---------------------------------------------------------------------------

#define NUM_CLASSES 19
#define NUM_SAMPLES 100
#define BATCH       8
#define HW          (256 * 256)
#define PIXELS      (BATCH * HW)          // 524288
#define BLOCK       256
#define GRID        (PIXELS / BLOCK)      // 2048

typedef float v2f __attribute__((ext_vector_type(2)));
typedef float v8f __attribute__((ext_vector_type(8)));

__device__ __forceinline__ unsigned hash_u32(unsigned x) {
    x ^= x >> 16; x *= 0x7FEB352Du;
    x ^= x >> 15; x *= 0x846CA68Bu;
    x ^= x >> 16;
    return x;
}

// Wave32 all-lane sum using V_WMMA_F32_16X16X4_F32.
// A (16x4) carries the 32 lane partials (lane l -> row l%16, K = 2*(l/16),
// second K slot zero), B = ones(4x16): D[m][n] = acc_m + acc_{m+16}.
// 7 adds over the 8 D VGPRs + one xor-16 shuffle give the full wave sum.
__device__ __forceinline__ float wave_reduce_wmma(float acc) {
    v2f a; a[0] = acc;  a[1] = 0.0f;
    v2f b; b[0] = 1.0f; b[1] = 1.0f;
    v8f c = {};
    v8f d = __builtin_amdgcn_wmma_f32_16x16x4_f32(
        false, a, false, b, (short)0, c, false, false);
    float h = ((d[0] + d[1]) + (d[2] + d[3])) + ((d[4] + d[5]) + (d[6] + d[7]));
    h += __shfl_xor(h, 16, 32);
    return h;
}

__global__ __launch_bounds__(BLOCK)
void hce_main_kernel(const float* __restrict__ inp,
                     const int*   __restrict__ target,
                     float*       __restrict__ block_sums) {
    const int pix  = blockIdx.x * BLOCK + threadIdx.x;   // grid covers PIXELS exactly
    const int b    = pix >> 16;                          // pix / HW
    const int p    = pix & (HW - 1);                     // pix % HW
    const long base = (long)b * (2 * NUM_CLASSES) * HW + p;
    const float* lg = inp + base;                          // logits planes
    const float* ls = inp + base + (long)NUM_CLASSES * HW; // log-std planes

    // gfx1250 global_prefetch_b8: warm the log-std planes while logits load.
    #pragma unroll
    for (int c = 0; c < NUM_CLASSES; c += 4)
        __builtin_prefetch(ls + (long)c * HW, 0, 0);

    // Per-pixel constants, log2 domain, CLT scale folded in:
    //   i ~ sum of 4 uniform bytes: mean 510, sigma 147.7985
    //   sample*log2e = i*sca[c] + off[c]
    const float L2E = 1.44269504f;
    const float ISD = 0.0067659811f;                     // 1/147.7985
    float sca[NUM_CLASSES], off[NUM_CLASSES];
    #pragma unroll
    for (int c = 0; c < NUM_CLASSES; c++) {
        float stdn = __expf(ls[(long)c * HW]);           // std = exp(log_std)
        float s    = stdn * (L2E * ISD);
        sca[c] = s;
        off[c] = fmaf(lg[(long)c * HW], L2E, -510.0f * s);
    }

    const int t = target[pix];
    const unsigned pseed = (unsigned)pix * 0x9E3779B1u;

    float acc2 = 0.0f;                                   // NLL accumulator in log2 units
    for (int s = 0; s < NUM_SAMPLES; s++) {
        unsigned st = hash_u32(pseed ^ ((unsigned)(s + 1) * 0x85EBCA6Bu));
        float sm[NUM_CLASSES];
        float mx  = -3.4e38f;
        float smt = 0.0f;
        #pragma unroll
        for (int c = 0; c < NUM_CLASSES; c++) {
            st = st * 1664525u + 1013904223u;            // LCG step
            // CLT-of-4 Gaussian: one V_DOT4_U32_U8 sums the 4 state bytes.
            unsigned i = __builtin_amdgcn_udot4(st, 0x01010101u, 0u, false);
            float v = fmaf((float)i, sca[c], off[c]);    // log2e * sampled logit
            sm[c] = v;
            mx = fmaxf(mx, v);
            if (c == t) smt = v;                         // hoisted-mask v_cndmask
        }
        float se = 0.0f;
        #pragma unroll
        for (int c = 0; c < NUM_CLASSES; c++)
            se += __builtin_amdgcn_exp2f(sm[c] - mx);    // raw v_exp_f32
        acc2 += (mx + __builtin_amdgcn_logf(se)) - smt;  // raw v_log_f32 (log2)
    }
    float acc = acc2 * 0.69314718056f;                   // back to natural log

    // Wave32 WMMA reduction, then LDS across the 8 waves of the block.
    __shared__ float ws[BLOCK / 32];
    float wsum = wave_reduce_wmma(acc);
    const int lane = threadIdx.x & 31;
    const int wid  = threadIdx.x >> 5;
    if (lane == 0) ws[wid] = wsum;
    __syncthreads();
    if (threadIdx.x == 0) {
        float bs = 0.0f;
        #pragma unroll
        for (int w = 0; w < BLOCK / 32; w++) bs += ws[w];
        block_sums[blockIdx.x] = bs;
    }
}

// Deterministic final reduction over the 2048 block partials (no float atomics).
__global__ __launch_bounds__(256)
void hce_finish_kernel(const float* __restrict__ block_sums,
                       float*       __restrict__ out) {
    float a = 0.0f;
    for (int i = threadIdx.x; i < GRID; i += 256) a += block_sums[i];
    __shared__ float ws[8];
    float w = wave_reduce_wmma(a);                       // EXEC all ones here
    if ((threadIdx.x & 31) == 0) ws[threadIdx.x >> 5] = w;
    __syncthreads();
    if (threadIdx.x == 0) {
        float tot = 0.0f;
        #pragma unroll
        for (int i = 0; i < 8; i++) tot += ws[i];
        out[0] = tot * (1.0f / ((float)PIXELS * (float)NUM_SAMPLES));
    }
}

extern "C" void kernel_launch(void* const* d_in, const int* in_sizes, int n_in,
                              void* d_out, int out_size, void* d_ws, size_t ws_size,
                              hipStream_t stream) {
    const float* inp    = (const float*)d_in[0];   // [8, 38, 256, 256] f32
    const int*   target = (const int*)d_in[1];     // [8, 256, 256] int32
    float* bsums = (float*)d_ws;                   // GRID floats of scratch

    hce_main_kernel<<<GRID, BLOCK, 0, stream>>>(inp, target, bsums);
    hce_finish_kernel<<<1, 256, 0, stream>>>(bsums, (float*)d_out);
}